// FlashMultiHeadLatentAttention_64579128262960
// MI455X (gfx1250) — compile-verified
//
#include <hip/hip_runtime.h>

// ---------------------------------------------------------------------------
// MLA forward for MI455X (gfx1250), all-fp32 via V_WMMA_F32_16X16X4_F32.
// Round 2: transposed LDS staging for B/V (fragment reads become single
// ds_load b64, no v_mov repacking) + register software-pipelining of the
// global->LDS staging so global-load latency hides behind WMMA compute.
// ---------------------------------------------------------------------------

typedef __attribute__((ext_vector_type(2))) float v2f;
typedef __attribute__((ext_vector_type(8))) float v8f;

__device__ __forceinline__ v8f wmma_f32_4(v2f a, v2f b, v8f c) {
  // (neg_a, A, neg_b, B, c_mod, C, reuse_a, reuse_b)
  return __builtin_amdgcn_wmma_f32_16x16x4_f32(false, a, false, b, (short)0, c,
                                               false, false);
}

#define HID 2048
#define HD 128
#define SCALE_QK 0.08838834764831843f  // 1/sqrt(128)

// ---------------------------------------------------------------------------
// Generic C[M,N] = A[M,K] @ B[K,N] + bias[N]    (row-major, dims % 128 == 0)
// 256 threads = 8 waves; 128x128 tile; BK=32; wave grid 4(m) x 2(n).
// A staged [m][k+2]; B staged TRANSPOSED [n][k+2] so B-fragments are
// contiguous float2 in LDS. Staging pipelined through registers.
// ---------------------------------------------------------------------------
__global__ __launch_bounds__(256) void gemm_bias_kernel(
    const float* __restrict__ A, const float* __restrict__ B,
    const float* __restrict__ bias, float* __restrict__ C,
    int M, int N, int K) {
  __shared__ float As[128][34];  // [m][k], pad 2
  __shared__ float Bs[128][34];  // [n][k], pad 2 (transposed)

  const int t = threadIdx.x;
  const int lane = t & 31;
  const int l16 = lane & 15;
  const int hf = lane >> 4;
  const int wave = t >> 5;
  const int wm = (wave >> 1) << 5;  // 0,32,64,96
  const int wn = (wave & 1) << 6;   // 0,64
  const int m0 = blockIdx.y << 7;
  const int n0 = blockIdx.x << 7;

  v8f acc[2][4] = {};
  float4 ra[4], rb[4];

  // Prologue: load tile kt=0 into registers.
#pragma unroll
  for (int p = 0; p < 4; ++p) {
    const int fl = t + (p << 8);
    ra[p] = *(const float4*)(A + (size_t)(m0 + (fl >> 3)) * K + ((fl & 7) << 2));
    rb[p] = *(const float4*)(B + (size_t)(fl >> 5) * N + n0 + ((fl & 31) << 2));
  }

  for (int kt = 0; kt < K; kt += 32) {
    // Store staged registers to LDS (A direct, B transposed).
#pragma unroll
    for (int p = 0; p < 4; ++p) {
      const int fl = t + (p << 8);
      const int rA = fl >> 3, cA = (fl & 7) << 2;
      *(v2f*)&As[rA][cA] = v2f{ra[p].x, ra[p].y};
      *(v2f*)&As[rA][cA + 2] = v2f{ra[p].z, ra[p].w};
      const int rB = fl >> 5, cB = (fl & 31) << 2;
      Bs[cB + 0][rB] = rb[p].x;
      Bs[cB + 1][rB] = rb[p].y;
      Bs[cB + 2][rB] = rb[p].z;
      Bs[cB + 3][rB] = rb[p].w;
    }
    __syncthreads();
    // Issue next tile's global loads; they complete while we do WMMA below.
    if (kt + 32 < K) {
#pragma unroll
      for (int p = 0; p < 4; ++p) {
        const int fl = t + (p << 8);
        ra[p] = *(const float4*)(A + (size_t)(m0 + (fl >> 3)) * K + kt + 32 +
                                 ((fl & 7) << 2));
        rb[p] = *(const float4*)(B + (size_t)(kt + 32 + (fl >> 5)) * N + n0 +
                                 ((fl & 31) << 2));
      }
    }
#pragma unroll
    for (int kk = 0; kk < 32; kk += 4) {
      const v2f a0 = *(const v2f*)&As[wm + l16][kk + 2 * hf];
      const v2f a1 = *(const v2f*)&As[wm + 16 + l16][kk + 2 * hf];
#pragma unroll
      for (int ni = 0; ni < 4; ++ni) {
        const v2f b = *(const v2f*)&Bs[wn + (ni << 4) + l16][kk + 2 * hf];
        acc[0][ni] = wmma_f32_4(a0, b, acc[0][ni]);
        acc[1][ni] = wmma_f32_4(a1, b, acc[1][ni]);
      }
    }
    __syncthreads();
  }

#pragma unroll
  for (int ni = 0; ni < 4; ++ni) {
    const int col = n0 + wn + (ni << 4) + l16;
    const float bv = bias[col];
#pragma unroll
    for (int mi = 0; mi < 2; ++mi) {
#pragma unroll
      for (int r = 0; r < 8; ++r) {
        const int row = m0 + wm + (mi << 4) + (hf << 3) + r;
        C[(size_t)row * N + col] = acc[mi][ni][r] + bv;
      }
    }
  }
}

// ---------------------------------------------------------------------------
// Blockwise-softmax attention (per-KV-block softmax, contexts summed).
// Grid: (qb=16, h=16). 256 threads = 8 waves; wave w owns q rows
// [qb*128 + 16w, +16). Staging register-pipelined across chunks AND phases:
// QK chunk 3 prefetches V chunk 0; PV chunk 3 prefetches next block's K.
// ---------------------------------------------------------------------------
__global__ __launch_bounds__(256) void mla_attention_kernel(
    const float* __restrict__ Q, const float* __restrict__ Kt,
    const float* __restrict__ V, float* __restrict__ ctx) {
  __shared__ float Ks[32][132];    // [kcol][hd] (+4 pad): frag = float2 row-run
  __shared__ float Vs[128][34];    // [hd][kcol] transposed: frag = float2
  __shared__ float Ps[8][16][34];  // per-wave P staging [q][kcol]

  const int t = threadIdx.x;
  const int lane = t & 31;
  const int l16 = lane & 15;
  const int hf = lane >> 4;
  const int w = t >> 5;
  const int qb = blockIdx.x;
  const int h = blockIdx.y;

  // Register-cache Q fragments: lane m=l16, K pair = 2*hf (+ 4*kk).
  v2f qf[32];
  {
    const float* qp =
        Q + (size_t)(qb * 128 + w * 16 + l16) * HID + h * HD + 2 * hf;
#pragma unroll
    for (int i = 0; i < 32; ++i) qf[i] = *(const v2f*)(qp + 4 * i);
  }

  v8f cacc[8] = {};
  float4 sreg[4];
  const int flr = t >> 5;                // staging row (0..31 per p-step of 8)
  // Per-p staging coords: fl = t + p*256 -> row = fl>>5, col4 = (fl&31)<<2.

  // Prologue: K chunk 0 of block 0.
#pragma unroll
  for (int p = 0; p < 4; ++p) {
    const int fl = t + (p << 8);
    sreg[p] = *(const float4*)(Kt + (size_t)(fl >> 5) * HID + h * HD +
                               ((fl & 31) << 2));
  }
  (void)flr;

  for (int kb = 0; kb < 16; ++kb) {
    v8f sacc[8] = {};
    // ---- S = Q @ K^T (K streamed in 4 chunks of 32 kv rows) ----
    for (int c = 0; c < 4; ++c) {
      __syncthreads();  // prior readers of Ks/Vs done
#pragma unroll
      for (int p = 0; p < 4; ++p) {
        const int fl = t + (p << 8);
        *(float4*)&Ks[fl >> 5][(fl & 31) << 2] = sreg[p];
      }
      __syncthreads();
      if (c < 3) {  // next K chunk
#pragma unroll
        for (int p = 0; p < 4; ++p) {
          const int fl = t + (p << 8);
          sreg[p] = *(const float4*)(
              Kt + (size_t)(kb * 128 + (c + 1) * 32 + (fl >> 5)) * HID +
              h * HD + ((fl & 31) << 2));
        }
      } else {  // prefetch V chunk 0 of this block
#pragma unroll
        for (int p = 0; p < 4; ++p) {
          const int fl = t + (p << 8);
          sreg[p] = *(const float4*)(V + (size_t)(kb * 128 + (fl >> 5)) * HID +
                                     h * HD + ((fl & 31) << 2));
        }
      }
#pragma unroll
      for (int kk = 0; kk < 128; kk += 4) {
        const v2f a = qf[kk >> 2];
#pragma unroll
        for (int ti = 0; ti < 2; ++ti) {
          const v2f b = *(const v2f*)&Ks[ti * 16 + l16][kk + 2 * hf];
          sacc[2 * c + ti] = wmma_f32_4(a, b, sacc[2 * c + ti]);
        }
      }
    }

    // ---- per-row softmax within this KV block (row = 8*hf + r) ----
#pragma unroll
    for (int r = 0; r < 8; ++r) {
      float m = -__builtin_inff();
#pragma unroll
      for (int ti = 0; ti < 8; ++ti) m = fmaxf(m, sacc[ti][r]);
      m = fmaxf(m, __shfl_xor(m, 1, 32));
      m = fmaxf(m, __shfl_xor(m, 2, 32));
      m = fmaxf(m, __shfl_xor(m, 4, 32));
      m = fmaxf(m, __shfl_xor(m, 8, 32));
      const float ms = m * SCALE_QK;
      float sum = 0.0f;
#pragma unroll
      for (int ti = 0; ti < 8; ++ti) {
        const float p = __expf(sacc[ti][r] * SCALE_QK - ms);
        sacc[ti][r] = p;
        sum += p;
      }
      sum += __shfl_xor(sum, 1, 32);
      sum += __shfl_xor(sum, 2, 32);
      sum += __shfl_xor(sum, 4, 32);
      sum += __shfl_xor(sum, 8, 32);
      const float inv = 1.0f / sum;
#pragma unroll
      for (int ti = 0; ti < 8; ++ti) sacc[ti][r] *= inv;
    }

    // ---- ctx += P @ V (V streamed in 4 chunks of 32 kv rows) ----
    for (int c = 0; c < 4; ++c) {
      __syncthreads();
      // Store V chunk transposed [hd][kcol] + stage this wave's P chunk.
#pragma unroll
      for (int p = 0; p < 4; ++p) {
        const int fl = t + (p << 8);
        const int r = fl >> 5, c4 = (fl & 31) << 2;
        Vs[c4 + 0][r] = sreg[p].x;
        Vs[c4 + 1][r] = sreg[p].y;
        Vs[c4 + 2][r] = sreg[p].z;
        Vs[c4 + 3][r] = sreg[p].w;
      }
#pragma unroll
      for (int ti = 0; ti < 2; ++ti)
#pragma unroll
        for (int r = 0; r < 8; ++r)
          Ps[w][8 * hf + r][ti * 16 + l16] = sacc[2 * c + ti][r];
      __syncthreads();
      if (c < 3) {  // next V chunk
#pragma unroll
        for (int p = 0; p < 4; ++p) {
          const int fl = t + (p << 8);
          sreg[p] = *(const float4*)(
              V + (size_t)(kb * 128 + (c + 1) * 32 + (fl >> 5)) * HID +
              h * HD + ((fl & 31) << 2));
        }
      } else if (kb < 15) {  // prefetch next block's K chunk 0
#pragma unroll
        for (int p = 0; p < 4; ++p) {
          const int fl = t + (p << 8);
          sreg[p] = *(const float4*)(
              Kt + (size_t)((kb + 1) * 128 + (fl >> 5)) * HID + h * HD +
              ((fl & 31) << 2));
        }
      }
#pragma unroll
      for (int kk = 0; kk < 32; kk += 4) {
        const v2f a = *(const v2f*)&Ps[w][l16][kk + 2 * hf];
#pragma unroll
        for (int nt = 0; nt < 8; ++nt) {
          const v2f b = *(const v2f*)&Vs[(nt << 4) + l16][kk + 2 * hf];
          cacc[nt] = wmma_f32_4(a, b, cacc[nt]);
        }
      }
    }
  }

  // ---- store ctx [S][HID] ----
#pragma unroll
  for (int nt = 0; nt < 8; ++nt) {
#pragma unroll
    for (int r = 0; r < 8; ++r) {
      const int row = qb * 128 + w * 16 + (hf << 3) + r;
      const int col = h * HD + (nt << 4) + l16;
      ctx[(size_t)row * HID + col] = cacc[nt][r];
    }
  }
}

// ---------------------------------------------------------------------------
extern "C" void kernel_launch(void* const* d_in, const int* in_sizes, int n_in,
                              void* d_out, int out_size, void* d_ws,
                              size_t ws_size, hipStream_t stream) {
  (void)in_sizes; (void)n_in; (void)out_size; (void)ws_size;
  const float* hs   = (const float*)d_in[0];
  const float* w_q  = (const float*)d_in[1];
  const float* b_q  = (const float*)d_in[2];
  const float* w_kv = (const float*)d_in[3];
  const float* b_kv = (const float*)d_in[4];
  const float* w_k  = (const float*)d_in[5];
  const float* b_k  = (const float*)d_in[6];
  const float* w_v  = (const float*)d_in[7];
  const float* b_v  = (const float*)d_in[8];
  const float* w_o  = (const float*)d_in[9];
  const float* b_o  = (const float*)d_in[10];
  float* out = (float*)d_out;

  float* q_ws = (float*)d_ws;                    // 2048*2048
  float* k_ws = q_ws + (size_t)2048 * 2048;      // 2048*2048
  float* v_ws = k_ws + (size_t)2048 * 2048;      // 2048*2048
  float* c_ws = v_ws + (size_t)2048 * 2048;      // 2048*2048
  float* l_ws = c_ws + (size_t)2048 * 2048;      // 2048*512

  const dim3 blk(256);
  gemm_bias_kernel<<<dim3(16, 16), blk, 0, stream>>>(hs, w_q, b_q, q_ws,
                                                     2048, 2048, 2048);
  gemm_bias_kernel<<<dim3(4, 16), blk, 0, stream>>>(hs, w_kv, b_kv, l_ws,
                                                    2048, 512, 2048);
  gemm_bias_kernel<<<dim3(16, 16), blk, 0, stream>>>(l_ws, w_k, b_k, k_ws,
                                                     2048, 2048, 512);
  gemm_bias_kernel<<<dim3(16, 16), blk, 0, stream>>>(l_ws, w_v, b_v, v_ws,
                                                     2048, 2048, 512);
  mla_attention_kernel<<<dim3(16, 16), blk, 0, stream>>>(q_ws, k_ws, v_ws,
                                                         c_ws);
  gemm_bias_kernel<<<dim3(16, 16), blk, 0, stream>>>(c_ws, w_o, b_o, out,
                                                     2048, 2048, 2048);
}